// Correlation_layer_54949811585132
// MI455X (gfx1250) — compile-verified
//
#include <hip/hip_runtime.h>

typedef __attribute__((ext_vector_type(2))) float v2f;
typedef __attribute__((ext_vector_type(8))) float v8f;

#define HH   96
#define WW   96
#define CC   128            // 64 + 64 concatenated channels
#define PADR 20
#define DD   41             // 2*pad+1
#define HW   (HH * WW)
#define WP   144            // padded width: 20 left + 96 + 28 right (covers all B tiles)
#define NJOBS (HH * DD * 6) // h * oy * w-tiles = 23616

// ---------------------------------------------------------------------------
// Prep 1: interleave two NCHW halves (64 ch each) into NHWC [h][w][c] (x1).
// ---------------------------------------------------------------------------
__global__ void prep_x1(const float* __restrict__ a0,
                        const float* __restrict__ a1,
                        float* __restrict__ dst) {
    int tid = blockIdx.x * blockDim.x + threadIdx.x;
    if (tid >= HW * CC) return;
    int c   = tid & (CC - 1);
    int pix = tid >> 7;
    float v = (c < 64) ? a0[c * HW + pix] : a1[(c - 64) * HW + pix];
    dst[tid] = v;   // dst[(h*96+w)*128 + c], coalesced writes
}

// ---------------------------------------------------------------------------
// Prep 2: interleave + PHYSICALLY ZERO-PAD x2 in W: dst[h][xp][c], xp=x+20,
// width 144. This removes all per-lane bounds masks (and the EXEC
// save/restore per load they caused) from the WMMA inner loop.
// ---------------------------------------------------------------------------
__global__ void prep_x2pad(const float* __restrict__ a0,
                           const float* __restrict__ a1,
                           float* __restrict__ dst) {
    int tid = blockIdx.x * blockDim.x + threadIdx.x;
    if (tid >= HH * WP * CC) return;
    int c    = tid & (CC - 1);
    int rest = tid >> 7;
    int xp   = rest % WP;
    int h    = rest / WP;
    int x    = xp - PADR;
    float v  = 0.0f;
    if (x >= 0 && x < WW) {
        int pix = h * WW + x;
        v = (c < 64) ? a0[c * HW + pix] : a1[(c - 64) * HW + pix];
    }
    dst[tid] = v;
}

// ---------------------------------------------------------------------------
// Correlation via V_WMMA_F32_16X16X4_F32.
// One wave = one (h, oy, 16-pixel w-tile) job.
//   A[m][k]   = x1t[h, w0+m, k]             (16 x 128)
//   B_t[k][n] = x2p[h2, 20 + w0-20+16t+n, k] = x2p[h2, w0+16t+n, k]
//   D_t[m][n] -> out[ oy*41 + (16t+n-m), h, w0+m ]  for 0 <= 16t+n-m <= 40
// t = floor((ox+m)/16) partitions every (m,ox) pair exactly once.
// Zero padding: W handled by the padded buffer, H by the uniform rowok skip.
// Fully unrolled K loop: all load addresses are base + immediate offset.
// ---------------------------------------------------------------------------
__global__ void corr_wmma(const float* __restrict__ x1t,
                          const float* __restrict__ x2p,
                          float* __restrict__ out) {
    const int wave = blockIdx.x * 8 + (threadIdx.x >> 5);
    const int lane = threadIdx.x & 31;
    if (wave >= NJOBS) return;

    const int w0t = wave % 6;
    int t1        = wave / 6;
    const int oy  = t1 % DD;
    const int h   = t1 / DD;
    const int w0  = w0t * 16;
    const int h2  = h + oy - PADR;
    const bool rowok = (h2 >= 0) && (h2 < HH);

    const int half = lane >> 4;      // 0: K={0,1}, 1: K={2,3}
    const int lm   = lane & 15;      // M (for A) / N (for B)
    const int kb   = half * 2;

    // A fragment base: x1t[(h*96 + w0 + m)*128 + kb]
    const float* aptr = x1t + ((size_t)(h * WW + w0 + lm) * CC + kb);

    v8f acc0 = {}, acc1 = {}, acc2 = {}, acc3 = {};

    if (rowok) {   // wave-uniform branch: EXEC stays all-ones for WMMA
        // B fragment bases in the padded buffer (always in-bounds)
        const size_t rowb = (size_t)h2 * WP;
        const float* b0p = x2p + ((rowb + (w0 +  0 + lm)) * CC + kb);
        const float* b1p = x2p + ((rowb + (w0 + 16 + lm)) * CC + kb);
        const float* b2p = x2p + ((rowb + (w0 + 32 + lm)) * CC + kb);
        const float* b3p = x2p + ((rowb + (w0 + 48 + lm)) * CC + kb);

#pragma unroll
        for (int k = 0; k < CC; k += 4) {
            v2f a  = *(const v2f*)(aptr + k);
            v2f b0 = *(const v2f*)(b0p + k);
            v2f b1 = *(const v2f*)(b1p + k);
            v2f b2 = *(const v2f*)(b2p + k);
            v2f b3 = *(const v2f*)(b3p + k);
            acc0 = __builtin_amdgcn_wmma_f32_16x16x4_f32(
                       false, a, false, b0, (short)0, acc0, false, false);
            acc1 = __builtin_amdgcn_wmma_f32_16x16x4_f32(
                       false, a, false, b1, (short)0, acc1, false, false);
            acc2 = __builtin_amdgcn_wmma_f32_16x16x4_f32(
                       false, a, false, b2, (short)0, acc2, false, false);
            acc3 = __builtin_amdgcn_wmma_f32_16x16x4_f32(
                       false, a, false, b3, (short)0, acc3, false, false);
        }
    }

    // Scatter D-tiles along displacement diagonals.
    // C/D layout: lane l, VGPR v -> M = v + 8*(l>>4), N = l & 15.
    const int obase = oy * DD;
#pragma unroll
    for (int v = 0; v < 8; ++v) {
        const int m = v + 8 * half;
        const int widx = h * WW + w0 + m;
        float vals[4] = {acc0[v], acc1[v], acc2[v], acc3[v]};
#pragma unroll
        for (int t = 0; t < 4; ++t) {
            int ox = 16 * t + lm - m;
            if ((unsigned)ox <= 40u) {
                __builtin_nontemporal_store(
                    vals[t], out + (size_t)(obase + ox) * HW + widx);
            }
        }
    }
}

// ---------------------------------------------------------------------------
// out_img: pure shifted-window gather of img (Ci = 1), streamed with NT stores.
// ---------------------------------------------------------------------------
__global__ void img_gather(const float* __restrict__ img,
                           float* __restrict__ out) {
    int idx = blockIdx.x * blockDim.x + threadIdx.x;
    if (idx >= DD * DD * HW) return;
    int w = idx % WW;
    int t = idx / WW;
    int h = t % HH;  t /= HH;
    int ox = t % DD;
    int oy = t / DD;
    int y = h + oy - PADR;
    int x = w + ox - PADR;
    float v = (y >= 0 && y < HH && x >= 0 && x < WW) ? img[y * WW + x] : 0.0f;
    __builtin_nontemporal_store(v, out + idx);
}

extern "C" void kernel_launch(void* const* d_in, const int* in_sizes, int n_in,
                              void* d_out, int out_size, void* d_ws, size_t ws_size,
                              hipStream_t stream) {
    const float* x_1 = (const float*)d_in[0];
    const float* x_2 = (const float*)d_in[1];
    const float* img = (const float*)d_in[2];
    const float* p_1 = (const float*)d_in[3];
    const float* p_2 = (const float*)d_in[4];
    float* out = (float*)d_out;

    float* x1t = (float*)d_ws;            // 96*96*128  floats = 4.5 MiB
    float* x2p = x1t + (size_t)HW * CC;   // 96*144*128 floats = 6.75 MiB

    const int n1 = HW * CC;
    prep_x1<<<(n1 + 255) / 256, 256, 0, stream>>>(x_1, p_1, x1t);

    const int n2 = HH * WP * CC;
    prep_x2pad<<<(n2 + 255) / 256, 256, 0, stream>>>(x_2, p_2, x2p);

    corr_wmma<<<NJOBS / 8, 256, 0, stream>>>(x1t, x2p, out);

    const int nimg = DD * DD * HW;
    img_gather<<<(nimg + 255) / 256, 256, 0, stream>>>(
        img, out + (size_t)DD * DD * HW);
}